// RoI2Det_88656714924159
// MI455X (gfx1250) — compile-verified
//
#include <hip/hip_runtime.h>
#include <hip/hip_bf16.h>

#define B_IMG 16
#define N_PROP 2048
#define N_CLS 80
#define NC1 81
#define MFLAT (N_PROP * N_CLS)          // 163840
#define MPAD (1 << 18)                  // 262144 (next pow2)
#define KTOP 2048
#define KW 64                           // KTOP/32
#define MAXPER 100
#define SCORE_THR 0.05f
#define MAX_RATIO 4.135166556742356f

typedef float v2f __attribute__((ext_vector_type(2)));
typedef float v8f __attribute__((ext_vector_type(8)));
typedef unsigned long long ull;

__device__ __forceinline__ unsigned f2key(float x) {
    unsigned u = __float_as_uint(x);
    return u ^ ((u >> 31) ? 0xFFFFFFFFu : 0x80000000u);
}
__device__ __forceinline__ float key2f(unsigned k) {
    unsigned u = (k & 0x80000000u) ? (k ^ 0x80000000u) : ~k;
    return __uint_as_float(u);
}

// ---------------- Kernel 1: softmax + decode + clip + sortable keys ----------
__global__ void k_decode(const float* __restrict__ cls, const float* __restrict__ reg,
                         const float* __restrict__ props, const int* __restrict__ hw,
                         ull* __restrict__ keys, float* __restrict__ boxesBuf) {
    int blk = blockIdx.x;             // B*N blocks
    int b = blk / N_PROP, p = blk % N_PROP;
    int t = threadIdx.x;              // 128 threads
    __shared__ float sl[128];
    __shared__ float red[128];

    float v = (t < NC1) ? cls[((long long)b * N_PROP + p) * NC1 + t] : -3.4e38f;
    sl[t] = v;
    red[t] = v;
    __syncthreads();
    for (int off = 64; off > 0; off >>= 1) {
        if (t < off) red[t] = fmaxf(red[t], red[t + off]);
        __syncthreads();
    }
    float mx = red[0];
    __syncthreads();
    red[t] = (t < NC1) ? __expf(sl[t] - mx) : 0.0f;
    __syncthreads();
    for (int off = 64; off > 0; off >>= 1) {
        if (t < off) red[t] += red[t + off];
        __syncthreads();
    }
    float denom = red[0];

    if (t < N_CLS) {
        float fg = __expf(sl[t] - mx) / denom;
        const float* pr = props + ((long long)b * N_PROP + p) * 4;
        float x1 = pr[0], y1 = pr[1], x2 = pr[2], y2 = pr[3];
        float pw = x2 - x1, ph = y2 - y1;
        float px = 0.5f * (x1 + x2), py = 0.5f * (y1 + y2);
        const float* d = reg + (((long long)b * N_PROP + p) * N_CLS + t) * 4;
        float dx = d[0] * 0.1f, dy = d[1] * 0.1f;
        float dw = fminf(fmaxf(d[2] * 0.2f, -MAX_RATIO), MAX_RATIO);
        float dh = fminf(fmaxf(d[3] * 0.2f, -MAX_RATIO), MAX_RATIO);
        float gw = pw * __expf(dw), gh = ph * __expf(dh);
        float gx = px + pw * dx, gy = py + ph * dy;
        float H = (float)hw[b * 2 + 0], W = (float)hw[b * 2 + 1];
        float bx0 = fminf(fmaxf(gx - 0.5f * gw, 0.0f), W);
        float by0 = fminf(fmaxf(gy - 0.5f * gh, 0.0f), H);
        float bx1 = fminf(fmaxf(gx + 0.5f * gw, 0.0f), W);
        float by1 = fminf(fmaxf(gy + 0.5f * gh, 0.0f), H);
        unsigned m = (unsigned)(p * N_CLS + t);
        float4* bb = (float4*)(boxesBuf + ((long long)b * MFLAT + m) * 4);
        *bb = make_float4(bx0, by0, bx1, by1);
        float masked = (fg > SCORE_THR) ? fg : -1.0f;
        keys[(long long)b * MPAD + m] =
            ((ull)f2key(masked) << 32) | (ull)(0xFFFFFFFFu - m);
    }
}

// ---------------- Kernel 1b: pad keys with smallest key ----------------------
__global__ void k_pad(ull* __restrict__ keys) {
    long long idx = (long long)blockIdx.x * blockDim.x + threadIdx.x;
    long long total = (long long)B_IMG * (MPAD - MFLAT);
    if (idx >= total) return;
    int b = (int)(idx / (MPAD - MFLAT));
    int i = (int)(idx % (MPAD - MFLAT));
    keys[(long long)b * MPAD + MFLAT + i] = 0ull;
}

// ---------------- Bitonic sort (descending) over each image's MPAD keys ------
__global__ void k_sort_local(ull* __restrict__ keys) {
    __shared__ ull s[2048];
    int t = threadIdx.x;                       // 1024
    long long base = (long long)blockIdx.x * 2048;
    s[t] = keys[base + t];
    s[t + 1024] = keys[base + t + 1024];
    __syncthreads();
    for (int k = 2; k <= 2048; k <<= 1) {
        for (int j = k >> 1; j > 0; j >>= 1) {
            int i = ((t & ~(j - 1)) << 1) | (t & (j - 1));
            int l = i | j;
            ull a = s[i], c = s[l];
            bool desc = (((unsigned)(base + i) & (unsigned)(MPAD - 1) & (unsigned)k) == 0);
            bool sw = desc ? (a < c) : (a > c);
            if (sw) { s[i] = c; s[l] = a; }
            __syncthreads();
        }
    }
    keys[base + t] = s[t];
    keys[base + t + 1024] = s[t + 1024];
}

__global__ void k_sort_global(ull* __restrict__ keys, int k, int j) {
    long long t = (long long)blockIdx.x * blockDim.x + threadIdx.x; // B*MPAD/2
    long long jm = j;
    long long i = ((t & ~(jm - 1)) << 1) | (t & (jm - 1));
    long long l = i | jm;
    ull a = keys[i], c = keys[l];
    bool desc = (((unsigned)i & (unsigned)(MPAD - 1) & (unsigned)k) == 0);
    bool sw = desc ? (a < c) : (a > c);
    if (sw) { keys[i] = c; keys[l] = a; }
}

__global__ void k_merge_local(ull* __restrict__ keys, int k) {
    __shared__ ull s[2048];
    int t = threadIdx.x;
    long long base = (long long)blockIdx.x * 2048;
    s[t] = keys[base + t];
    s[t + 1024] = keys[base + t + 1024];
    __syncthreads();
    bool desc = (((unsigned)base & (unsigned)(MPAD - 1) & (unsigned)k) == 0);
    for (int j = 1024; j > 0; j >>= 1) {
        int i = ((t & ~(j - 1)) << 1) | (t & (j - 1));
        int l = i | j;
        ull a = s[i], c = s[l];
        bool sw = desc ? (a < c) : (a > c);
        if (sw) { s[i] = c; s[l] = a; }
        __syncthreads();
    }
    keys[base + t] = s[t];
    keys[base + t + 1024] = s[t + 1024];
}

// ---------------- Kernel 3: gather top-K, build offset boxes -----------------
__global__ void k_gather(const ull* __restrict__ keys, const float* __restrict__ boxesBuf,
                         const int* __restrict__ hw, float* __restrict__ topBox,
                         float* __restrict__ offBox, float* __restrict__ topScore,
                         int* __restrict__ topLabel, unsigned* __restrict__ validBuf) {
    int t = blockIdx.x * blockDim.x + threadIdx.x;  // B*KTOP
    int b = t / KTOP, r = t % KTOP;
    ull key = keys[(long long)b * MPAD + r];
    unsigned m = 0xFFFFFFFFu - (unsigned)(key & 0xFFFFFFFFull);
    float s = key2f((unsigned)(key >> 32));
    int label = (int)(m % N_CLS);
    const float4* bb = (const float4*)(boxesBuf + ((long long)b * MFLAT + m) * 4);
    float4 bx = *bb;
    float4* tb = (float4*)(topBox + ((long long)b * KTOP + r) * 4);
    *tb = bx;
    float W = (float)hw[b * 2 + 1], H = (float)hw[b * 2 + 0];
    float off = (float)label * (fmaxf(W, H) + 1.0f);
    float4* ob = (float4*)(offBox + ((long long)b * KTOP + r) * 4);
    *ob = make_float4(bx.x + off, bx.y + off, bx.z + off, bx.w + off);
    topScore[b * KTOP + r] = s;
    topLabel[b * KTOP + r] = label;
    validBuf[b * KTOP + r] = (s > SCORE_THR) ? 1u : 0u;
}

// ---------------- Kernel 4: pairwise IoU -> suppression bitmask --------------
// One wave per 16-row x 32-col tile. Union base (area_i + area_j) via
// v_wmma_f32_16x16x4_f32 rank-2 outer sum; per-lane VALU for intersection;
// ballot_w32 packs predicates straight into bitmask words.
__global__ void k_iou(const float* __restrict__ offBox, unsigned* __restrict__ supMask) {
    int lane = threadIdx.x & 31;
    int wib = threadIdx.x >> 5;                 // 8 waves / block
    int gw = blockIdx.x * 8 + wib;              // B * 128 * 64 waves total
    int b = gw / (128 * 64);
    int rem = gw % (128 * 64);
    int ti = rem / 64, cw = rem % 64;
    int r0 = ti * 16, c0 = cw * 32;

    __shared__ float rowsS[8][16][4];
    __shared__ float colsS[8][32][4];
    __shared__ unsigned balS[8][2][8];

    const float4* base4 = (const float4*)(offBox + (long long)b * KTOP * 4);
    if (lane < 16) {
        float4 rv = base4[r0 + lane];
        rowsS[wib][lane][0] = rv.x; rowsS[wib][lane][1] = rv.y;
        rowsS[wib][lane][2] = rv.z; rowsS[wib][lane][3] = rv.w;
    }
    float4 cv = base4[c0 + lane];
    colsS[wib][lane][0] = cv.x; colsS[wib][lane][1] = cv.y;
    colsS[wib][lane][2] = cv.z; colsS[wib][lane][3] = cv.w;
    __syncthreads();

    float aRow = 0.0f;
    if (lane < 16) {
        const float* rb = rowsS[wib][lane];
        aRow = fmaxf(rb[2] - rb[0], 0.0f) * fmaxf(rb[3] - rb[1], 0.0f);
    }
    int n = lane & 15;
    float aColH[2];
#pragma unroll
    for (int h = 0; h < 2; h++) {
        const float* cb = colsS[wib][h * 16 + n];
        aColH[h] = fmaxf(cb[2] - cb[0], 0.0f) * fmaxf(cb[3] - cb[1], 0.0f);
    }

    // Union base tile: D[m][n] = area_row[m]*1 + 1*area_col[n]  (rank-2 WMMA)
    v8f cU[2];
#pragma unroll
    for (int h = 0; h < 2; h++) {
        v2f a, bv;
        a.x = (lane < 16) ? aRow : 0.0f;     // A[m][0] = area_row[m]
        a.y = (lane < 16) ? 1.0f : 0.0f;     // A[m][1] = 1
        bv.x = (lane < 16) ? 1.0f : 0.0f;    // B[0][n] = 1
        bv.y = (lane < 16) ? aColH[h] : 0.0f;// B[1][n] = area_col[n]
        v8f z = {0.f, 0.f, 0.f, 0.f, 0.f, 0.f, 0.f, 0.f};
        cU[h] = __builtin_amdgcn_wmma_f32_16x16x4_f32(
            false, a, false, bv, (short)0, z, false, false);
    }

    int mbase = (lane >= 16) ? 8 : 0;
#pragma unroll
    for (int h = 0; h < 2; h++) {
        const float* cb = colsS[wib][h * 16 + n];
        float cx0 = cb[0], cy0 = cb[1], cx1 = cb[2], cy1 = cb[3];
#pragma unroll
        for (int v = 0; v < 8; v++) {
            const float* rb = rowsS[wib][mbase + v];
            float iw = fminf(rb[2], cx1) - fmaxf(rb[0], cx0);
            float ih = fminf(rb[3], cy1) - fmaxf(rb[1], cy0);
            float inter = fmaxf(iw, 0.0f) * fmaxf(ih, 0.0f);
            float uni = cU[h][v] - inter;
            bool pred = inter > 0.5f * uni;   // iou > 0.5 (uni >= inter >= 0)
            unsigned bal = __builtin_amdgcn_ballot_w32(pred);
            if (lane == 0) balS[wib][h][v] = bal;
        }
    }
    __syncthreads();

    if (lane < 16) {
        int v = lane & 7;
        bool hi = lane >= 8;
        unsigned b0 = balS[wib][0][v], b1 = balS[wib][1][v];
        unsigned w = hi ? ((b0 >> 16) | (b1 & 0xFFFF0000u))
                        : ((b0 & 0xFFFFu) | (b1 << 16));
        supMask[((long long)b * KTOP + r0 + lane) * KW + cw] = w;
    }
}

// ---------------- Kernel 5: serial greedy NMS (one wave per image) -----------
__global__ void k_nms(const unsigned* __restrict__ supMask,
                      const unsigned* __restrict__ validBuf,
                      unsigned* __restrict__ keepBuf) {
    int b = blockIdx.x;
    int lane = threadIdx.x;                   // 32
    unsigned k0 = 0, k1 = 0;
    for (int bit = 0; bit < 32; bit++) {
        k0 |= (validBuf[b * KTOP + lane * 32 + bit] & 1u) << bit;
        k1 |= (validBuf[b * KTOP + (lane + 32) * 32 + bit] & 1u) << bit;
    }
    const unsigned* sm = supMask + (long long)b * KTOP * KW;
    for (int i = 0; i < KTOP; i++) {
        int wi = i >> 5, bt = i & 31;
        unsigned kw = (wi < 32) ? (unsigned)__shfl((int)k0, wi)
                                : (unsigned)__shfl((int)k1, wi - 32);
        unsigned alive = (kw >> bt) & 1u;
        const unsigned* row = sm + (long long)i * KW;
        unsigned s0 = row[lane], s1 = row[lane + 32];
        if (i + 4 < KTOP)
            __builtin_prefetch(row + 4 * KW + lane, 0, 0); // global_prefetch_b8
        unsigned tail = (bt == 31) ? 0u : (0xFFFFFFFFu << (bt + 1));
        unsigned m0 = (lane < wi) ? 0u : ((lane == wi) ? (s0 & tail) : s0);
        int l32 = lane + 32;
        unsigned m1 = (l32 < wi) ? 0u : ((l32 == wi) ? (s1 & tail) : s1);
        unsigned am = (unsigned)(-(int)alive);
        k0 &= ~(m0 & am);
        k1 &= ~(m1 & am);
    }
    keepBuf[b * KW + lane] = k0;
    keepBuf[b * KW + lane + 32] = k1;
}

// ---------------- Kernel 6: final sorted top-100 + outputs -------------------
__global__ void k_final(const unsigned* __restrict__ keepBuf,
                        const float* __restrict__ topScore,
                        const float* __restrict__ topBox,
                        const int* __restrict__ topLabel,
                        float* __restrict__ out) {
    __shared__ ull s[2048];
    int b = blockIdx.x;
    int t = threadIdx.x;                      // 1024
#pragma unroll
    for (int q = 0; q < 2; q++) {
        int r = t + q * 1024;
        unsigned kp = (keepBuf[b * KW + (r >> 5)] >> (r & 31)) & 1u;
        float sc = kp ? topScore[b * KTOP + r] : -1.0f;
        s[r] = ((ull)f2key(sc) << 32) | (ull)(0xFFFFFFFFu - (unsigned)r);
    }
    __syncthreads();
    for (int k = 2; k <= 2048; k <<= 1) {
        for (int j = k >> 1; j > 0; j >>= 1) {
            int i = ((t & ~(j - 1)) << 1) | (t & (j - 1));
            int l = i | j;
            ull a = s[i], c = s[l];
            bool desc = ((i & k) == 0);
            bool sw = desc ? (a < c) : (a > c);
            if (sw) { s[i] = c; s[l] = a; }
            __syncthreads();
        }
    }
    if (t < MAXPER) {
        ull key = s[t];
        unsigned r = 0xFFFFFFFFu - (unsigned)(key & 0xFFFFFFFFull);
        float sc = key2f((unsigned)(key >> 32));
        bool good = sc > 0.0f;
        float* det_boxes = out;                                   // [16,100,4]
        float* det_scores = out + B_IMG * MAXPER * 4;             // [16,100]
        float* det_labels = out + B_IMG * MAXPER * 4 + B_IMG * MAXPER;
        const float* bx = topBox + ((long long)b * KTOP + r) * 4;
        int o = b * MAXPER + t;
#pragma unroll
        for (int c = 0; c < 4; c++)
            det_boxes[o * 4 + c] = good ? bx[c] : 0.0f;
        det_scores[o] = good ? sc : 0.0f;
        det_labels[o] = good ? (float)topLabel[b * KTOP + r] : -1.0f;
    }
}

extern "C" void kernel_launch(void* const* d_in, const int* in_sizes, int n_in,
                              void* d_out, int out_size, void* d_ws, size_t ws_size,
                              hipStream_t stream) {
    const float* cls = (const float*)d_in[0];
    const float* reg = (const float*)d_in[1];
    const float* props = (const float*)d_in[2];
    const int* hw = (const int*)d_in[3];
    float* out = (float*)d_out;

    char* w = (char*)d_ws;
    size_t off = 0;
    ull* keys = (ull*)(w + off); off += (size_t)B_IMG * MPAD * 8;          // 32 MB
    size_t boxOff = off;
    float* boxesBuf = (float*)(w + off); off += (size_t)B_IMG * MFLAT * 16;// 40 MB
    unsigned* supMask = (unsigned*)(w + boxOff);  // aliases boxesBuf (used after)
    float* topBox = (float*)(w + off); off += (size_t)B_IMG * KTOP * 16;
    float* offBox = (float*)(w + off); off += (size_t)B_IMG * KTOP * 16;
    float* topScore = (float*)(w + off); off += (size_t)B_IMG * KTOP * 4;
    int* topLabel = (int*)(w + off); off += (size_t)B_IMG * KTOP * 4;
    unsigned* validBuf = (unsigned*)(w + off); off += (size_t)B_IMG * KTOP * 4;
    unsigned* keepBuf = (unsigned*)(w + off); off += (size_t)B_IMG * KW * 4;

    // 1) softmax + decode + keys
    k_decode<<<B_IMG * N_PROP, 128, 0, stream>>>(cls, reg, props, hw, keys, boxesBuf);
    // 1b) pad
    {
        long long total = (long long)B_IMG * (MPAD - MFLAT);
        k_pad<<<(unsigned)((total + 255) / 256), 256, 0, stream>>>(keys);
    }
    // 2) per-image descending bitonic sort of MPAD 64-bit keys
    k_sort_local<<<B_IMG * MPAD / 2048, 1024, 0, stream>>>(keys);
    for (int k = 4096; k <= MPAD; k <<= 1) {
        for (int j = k >> 1; j >= 2048; j >>= 1) {
            k_sort_global<<<(B_IMG * MPAD / 2) / 256, 256, 0, stream>>>(keys, k, j);
        }
        k_merge_local<<<B_IMG * MPAD / 2048, 1024, 0, stream>>>(keys, k);
    }
    // 3) gather top-K
    k_gather<<<(B_IMG * KTOP) / 256, 256, 0, stream>>>(
        keys, boxesBuf, hw, topBox, offBox, topScore, topLabel, validBuf);
    // 4) IoU -> suppression bitmask (WMMA union base + ballot pack)
    k_iou<<<(B_IMG * 128 * 64) / 8, 256, 0, stream>>>(offBox, supMask);
    // 5) serial greedy NMS, one wave per image
    k_nms<<<B_IMG, 32, 0, stream>>>(supMask, validBuf, keepBuf);
    // 6) final sorted top-100 + outputs
    k_final<<<B_IMG, 1024, 0, stream>>>(keepBuf, topScore, topBox, topLabel, out);
}